// SelfconLoss_21603685499217
// MI455X (gfx1250) — compile-verified
//
#include <hip/hip_runtime.h>
#include <math.h>

typedef float v2f __attribute__((ext_vector_type(2)));
typedef float v8f __attribute__((ext_vector_type(8)));

#define N_ROWS   65536
#define D        512
#define LAM      0.01f

#define NBLK_A   512
#define ROWS_PER_BLK (N_ROWS / NBLK_A)   // 128
#define TPB_A    128                     // each thread: one float4 column group

#define NBLK_B   128
#define TPB_B    256                     // 8 waves -> 8 tiles/block, 1024 tiles total

// ---------------------------------------------------------------------------
// Kernel 1: streaming reduction over z_i / z_j. Memory-bound (268 MB @ 23.3TB/s).
// Produces per-block column-sum partials + block-reduced dot / uu partials.
// ---------------------------------------------------------------------------
__global__ void __launch_bounds__(TPB_A)
selfcon_reduce_kernel(const float* __restrict__ zi, const float* __restrict__ zj,
                      float* __restrict__ ws_col,   // [NBLK_A * D]
                      float* __restrict__ ws_dot,   // [NBLK_A]
                      float* __restrict__ ws_uu)    // [NBLK_A]
{
    const int b = blockIdx.x;
    const int t = threadIdx.x;                 // 0..127 -> float4 column group
    const long rowBase = (long)b * ROWS_PER_BLK;
    const int stride4 = D / 4;                 // 128 float4 per row

    const float4* zi4 = (const float4*)zi;
    const float4* zj4 = (const float4*)zj;

    float4 colsum = make_float4(0.f, 0.f, 0.f, 0.f);
    float dot = 0.f, uu = 0.f;

    for (int r = 0; r < ROWS_PER_BLK; ++r) {
        long idx = (rowBase + r) * stride4 + t;
        float4 a = zi4[idx];
        float4 c = zj4[idx];
        float4 u = make_float4(a.x + c.x, a.y + c.y, a.z + c.z, a.w + c.w);
        dot += a.x * c.x + a.y * c.y + a.z * c.z + a.w * c.w;
        uu  += u.x * u.x + u.y * u.y + u.z * u.z + u.w * u.w;
        colsum.x += u.x; colsum.y += u.y; colsum.z += u.z; colsum.w += u.w;
    }

    ((float4*)ws_col)[b * stride4 + t] = colsum;

    __shared__ float sdot[TPB_A];
    __shared__ float suu[TPB_A];
    sdot[t] = dot; suu[t] = uu;
    __syncthreads();
    for (int s = TPB_A / 2; s > 0; s >>= 1) {
        if (t < s) { sdot[t] += sdot[t + s]; suu[t] += suu[t + s]; }
        __syncthreads();
    }
    if (t == 0) { ws_dot[b] = sdot[0]; ws_uu[b] = suu[0]; }
}

// ---------------------------------------------------------------------------
// Kernel 2: WW = W @ W^T via V_WMMA_F32_16X16X4_F32 (full f32 precision).
// One 16x16 WW tile per wave; accumulate sum of WW^4 over the tile.
// ISA layout (16x4 f32 A): lane L -> M = L%16; VGPR v -> K = v + 2*(L/16).
// B (4x16, = W^T tile) uses the mirrored layout -> identical float2 fetch
// pattern with the tile-column base swapped in. EXEC is all-ones (no
// divergence around the WMMA), as required.
// ---------------------------------------------------------------------------
__global__ void __launch_bounds__(TPB_B)
selfcon_wwt_kernel(const float* __restrict__ W, float* __restrict__ ws_s4)
{
    const int t    = threadIdx.x;
    const int wave = t >> 5;
    const int lane = t & 31;
    const int tile = blockIdx.x * (TPB_B / 32) + wave;   // 0..1023
    const int tm   = (tile >> 5) << 4;                   // WW tile row base
    const int tn   = (tile & 31) << 4;                   // WW tile col base
    const int mr   = lane & 15;
    const int koff = (lane >> 4) << 1;                   // 0 or 2

    const float2* Wa = (const float2*)(W + (long)(tm + mr) * D);
    const float2* Wb = (const float2*)(W + (long)(tn + mr) * D);

    v8f c = {0.f, 0.f, 0.f, 0.f, 0.f, 0.f, 0.f, 0.f};

    for (int k0 = 0; k0 < D; k0 += 4) {
        float2 af = Wa[(k0 + koff) >> 1];
        float2 bf = Wb[(k0 + koff) >> 1];
        v2f a;  a[0]  = af.x; a[1]  = af.y;
        v2f b2; b2[0] = bf.x; b2[1] = bf.y;
        // 8 args: (neg_a, A, neg_b, B, c_mod, C, reuse_a, reuse_b)
        c = __builtin_amdgcn_wmma_f32_16x16x4_f32(
                false, a, false, b2, (short)0, c, false, false);
    }

    float s4 = 0.f;
#pragma unroll
    for (int r = 0; r < 8; ++r) { float q = c[r] * c[r]; s4 += q * q; }

    __shared__ float sred[TPB_B];
    sred[t] = s4;
    __syncthreads();
    for (int s = TPB_B / 2; s > 0; s >>= 1) {
        if (t < s) sred[t] += sred[t + s];
        __syncthreads();
    }
    if (t == 0) ws_s4[blockIdx.x] = sred[0];
}

// ---------------------------------------------------------------------------
// Kernel 3: deterministic finalize. col dot col, sum partials, scalar math.
// ---------------------------------------------------------------------------
__global__ void __launch_bounds__(512)
selfcon_final_kernel(const float* __restrict__ ws_col, const float* __restrict__ ws_dot,
                     const float* __restrict__ ws_uu,  const float* __restrict__ ws_s4,
                     float* __restrict__ out)
{
    const int t = threadIdx.x;     // 0..511 (one per column)

    float col = 0.f;
    for (int b = 0; b < NBLK_A; ++b) col += ws_col[b * D + t];
    float colsq = col * col;

    float dotp = ws_dot[t];                       // NBLK_A == 512 == blockDim
    float uup  = ws_uu[t];
    float s4p  = (t < NBLK_B) ? ws_s4[t] : 0.f;

    __shared__ float4 sred[512];
    sred[t] = make_float4(colsq, dotp, uup, s4p);
    __syncthreads();
    for (int s = 256; s > 0; s >>= 1) {
        if (t < s) {
            float4 a = sred[t], b = sred[t + s];
            sred[t] = make_float4(a.x + b.x, a.y + b.y, a.z + b.z, a.w + b.w);
        }
        __syncthreads();
    }
    if (t == 0) {
        float total = sred[0].x;   // sum of all entries of u @ u^T
        float dot   = sred[0].y;   // sum z_i * z_j
        float uu    = sred[0].z;   // trace of u @ u^T
        float s4    = sred[0].w;   // sum (W W^T)^4

        float n       = (float)N_ROWS;
        float pos     = 2.f * dot;
        float offdiag = total - uu;
        float loss    = -(pos - offdiag / (2.f * n - 2.f)) / (2.f * n);
        float reg     = 0.5f * LAM * sqrtf(s4);
        out[0] = loss + reg;
    }
}

// ---------------------------------------------------------------------------
extern "C" void kernel_launch(void* const* d_in, const int* in_sizes, int n_in,
                              void* d_out, int out_size, void* d_ws, size_t ws_size,
                              hipStream_t stream)
{
    const float* zi = (const float*)d_in[0];   // [65536, 512] f32
    const float* zj = (const float*)d_in[1];   // [65536, 512] f32
    const float* W  = (const float*)d_in[2];   // [512, 512]   f32
    float* out = (float*)d_out;

    float* ws_col = (float*)d_ws;                       // NBLK_A * D
    float* ws_dot = ws_col + (size_t)NBLK_A * D;        // NBLK_A
    float* ws_uu  = ws_dot + NBLK_A;                    // NBLK_A
    float* ws_s4  = ws_uu  + NBLK_A;                    // NBLK_B

    selfcon_reduce_kernel<<<NBLK_A, TPB_A, 0, stream>>>(zi, zj, ws_col, ws_dot, ws_uu);
    selfcon_wwt_kernel<<<NBLK_B, TPB_B, 0, stream>>>(W, ws_s4);
    selfcon_final_kernel<<<1, 512, 0, stream>>>(ws_col, ws_dot, ws_uu, ws_s4, out);
}